// HeteroGCN_6004364280319
// MI455X (gfx1250) — compile-verified
//
#include <hip/hip_runtime.h>
#include <math.h>

typedef float v2f __attribute__((ext_vector_type(2)));
typedef float v8f __attribute__((ext_vector_type(8)));

#define NN 50000
#define NE 800000
#define NP 200000
#define HASH_BITS 21
#define HASH_CAP (1u << HASH_BITS)
#define HASH_MASK (HASH_CAP - 1u)

// ---- order-preserving float<->uint encoding for atomic segment-max ----
__device__ __forceinline__ unsigned enc_f32(float f) {
  unsigned u = __float_as_uint(f);
  return (u & 0x80000000u) ? ~u : (u | 0x80000000u);
}
__device__ __forceinline__ float dec_f32(unsigned u) {
  return (u & 0x80000000u) ? __uint_as_float(u & 0x7FFFFFFFu)
                           : __uint_as_float(~u);
}

// ---- Z[M x 64] = X[M x K] @ W[K x 64], fp32 WMMA, one wave per 16x16 tile ----
template <int K>
__global__ __launch_bounds__(128) void gemm_node_wmma(
    const float* __restrict__ X, const float* __restrict__ W,
    float* __restrict__ Z) {
  __shared__ float Ws[K * 64];
  __shared__ float As[16 * K];
  const int tid  = threadIdx.x;
  const int wave = tid >> 5;        // 4 waves -> 4 column tiles of 16
  const int lane = tid & 31;
  const int row0 = blockIdx.x * 16; // 16 node rows per block (50000 % 16 == 0)

  for (int i = tid; i < K * 64; i += 128) Ws[i] = W[i];
  for (int i = tid; i < 16 * K; i += 128) {
    int r = i / K, c = i % K;
    As[i] = X[(size_t)(row0 + r) * K + c];
  }
  __syncthreads();

  const int n0   = wave * 16;
  const int m    = lane & 15;   // M index for A, N index for B/D
  const int half = lane >> 4;   // K-half selector per ISA layout

  v8f c = {};
  for (int kk = 0; kk < K; kk += 4) {
    const int ka = kk + half * 2;
    v2f a, b;
    a.x = As[m * K + ka];
    a.y = As[m * K + ka + 1];
    b.x = Ws[ka * 64 + n0 + m];
    b.y = Ws[(ka + 1) * 64 + n0 + m];
    c = __builtin_amdgcn_wmma_f32_16x16x4_f32(false, a, false, b, (short)0, c,
                                              false, false);
  }
  // D layout: VGPR r -> M=r (lanes 0-15) / M=r+8 (lanes 16-31), N=lane&15
  for (int r = 0; r < 8; ++r) {
    const int mo = r + half * 8;
    Z[(size_t)(row0 + mo) * 64 + n0 + m] = c[r];
  }
}

// ---- el/er = Z . a_l / a_r, one wave per node ----
__global__ __launch_bounds__(256) void elr_kernel(
    const float* __restrict__ Z, const float* __restrict__ al,
    const float* __restrict__ ar, float* __restrict__ el,
    float* __restrict__ er) {
  const int node = (int)((blockIdx.x * blockDim.x + threadIdx.x) >> 5);
  const int lane = threadIdx.x & 31;
  if (node >= NN) return;
  const float z0 = Z[(size_t)node * 64 + lane];
  const float z1 = Z[(size_t)node * 64 + lane + 32];
  float sl = z0 * al[lane] + z1 * al[lane + 32];
  float sr = z0 * ar[lane] + z1 * ar[lane + 32];
  for (int o = 16; o; o >>= 1) {
    sl += __shfl_xor(sl, o, 32);
    sr += __shfl_xor(sr, o, 32);
  }
  if (lane == 0) { el[node] = sl; er[node] = sr; }
}

// ---- pass A: e = leaky_relu(el[src]+er[dst]); segment-max over dst ----
__global__ void edge_attn_max(const int* __restrict__ src,
                              const int* __restrict__ dst,
                              const float* __restrict__ el,
                              const float* __restrict__ er,
                              float* __restrict__ e_out,
                              unsigned* __restrict__ emax_enc) {
  const int i = blockIdx.x * blockDim.x + threadIdx.x;
  if (i >= NE) return;
  const int s = src[i], d = dst[i];
  const float x = el[s] + er[d];
  const float e = x > 0.f ? x : 0.2f * x;
  e_out[i] = e;
  atomicMax(&emax_enc[d], enc_f32(e));
}

// ---- pass B: ex = exp(e - max); denom[dst]+=ex; agg[dst] += ex*z[src] ----
// one wave per edge, 2 channels per lane
__global__ __launch_bounds__(256) void edge_aggregate(
    const int* __restrict__ src, const int* __restrict__ dst,
    const float* __restrict__ e_edge, const unsigned* __restrict__ emax_enc,
    const float* __restrict__ Z, float* __restrict__ denom,
    float* __restrict__ agg) {
  const int edge = (int)((blockIdx.x * blockDim.x + threadIdx.x) >> 5);
  const int lane = threadIdx.x & 31;
  if (edge >= NE) return;
  const int s = src[edge], d = dst[edge];
  const float ex = expf(e_edge[edge] - dec_f32(emax_enc[d]));
  if (lane == 0) atomicAdd(&denom[d], ex);
  const float v0 = Z[(size_t)s * 64 + lane];
  const float v1 = Z[(size_t)s * 64 + lane + 32];
  atomicAdd(&agg[(size_t)d * 64 + lane], ex * v0);
  atomicAdd(&agg[(size_t)d * 64 + lane + 32], ex * v1);
}

// ---- H = relu(agg / max(denom, 1e-9)) ----
__global__ void norm_relu(const float* __restrict__ agg,
                          const float* __restrict__ denom,
                          float* __restrict__ H) {
  const int i = blockIdx.x * blockDim.x + threadIdx.x;
  if (i >= NN * 64) return;
  const float dn = fmaxf(denom[i >> 6], 1e-9f);
  const float v  = agg[i] / dn;
  H[i] = v > 0.f ? v : 0.f;
}

// ---- hash set of edge keys for the existence mask ----
__device__ __forceinline__ unsigned hash_slot(unsigned long long key) {
  return (unsigned)((key * 0x9E3779B97F4A7C15ull) >> (64 - HASH_BITS));
}

__global__ void hash_insert(const int* __restrict__ src,
                            const int* __restrict__ dst,
                            unsigned long long* __restrict__ tab) {
  const int i = blockIdx.x * blockDim.x + threadIdx.x;
  if (i >= NE) return;
  const unsigned long long key =
      (unsigned long long)((long long)src[i] * NN + dst[i]) + 1ull;
  unsigned idx = hash_slot(key) & HASH_MASK;
  for (;;) {
    unsigned long long old = atomicCAS(&tab[idx], 0ull, key);
    if (old == 0ull || old == key) break;
    idx = (idx + 1u) & HASH_MASK;
  }
}

// ---- pair head: sigmoid([h_s, h_d] @ Wc + bc) * exists(s,d) ----
__global__ __launch_bounds__(256) void pair_score(
    const float* __restrict__ H, const float* __restrict__ Wc,
    const float* __restrict__ bc, const int* __restrict__ psrc,
    const int* __restrict__ pdst, const unsigned long long* __restrict__ tab,
    float* __restrict__ out) {
  const int p = (int)((blockIdx.x * blockDim.x + threadIdx.x) >> 5);
  const int lane = threadIdx.x & 31;
  if (p >= NP) return;
  const int s = psrc[p], d = pdst[p];
  float acc = H[(size_t)s * 64 + lane] * Wc[lane] +
              H[(size_t)s * 64 + lane + 32] * Wc[lane + 32] +
              H[(size_t)d * 64 + lane] * Wc[64 + lane] +
              H[(size_t)d * 64 + lane + 32] * Wc[96 + lane];
  for (int o = 16; o; o >>= 1) acc += __shfl_xor(acc, o, 32);
  if (lane == 0) {
    const float sig = 1.f / (1.f + expf(-(acc + bc[0])));
    const unsigned long long key =
        (unsigned long long)((long long)s * NN + d) + 1ull;
    unsigned idx = hash_slot(key) & HASH_MASK;
    float mask = 0.f;
    for (;;) {
      const unsigned long long v = tab[idx];
      if (v == key) { mask = 1.f; break; }
      if (v == 0ull) break;
      idx = (idx + 1u) & HASH_MASK;
    }
    out[p] = sig * mask;
  }
}

extern "C" void kernel_launch(void* const* d_in, const int* in_sizes, int n_in,
                              void* d_out, int out_size, void* d_ws,
                              size_t ws_size, hipStream_t stream) {
  const float* feat = (const float*)d_in[0];
  const float* W1   = (const float*)d_in[1];
  const float* al1  = (const float*)d_in[2];
  const float* ar1  = (const float*)d_in[3];
  const float* W2   = (const float*)d_in[4];
  const float* al2  = (const float*)d_in[5];
  const float* ar2  = (const float*)d_in[6];
  const float* Wc   = (const float*)d_in[7];
  const float* bc   = (const float*)d_in[8];
  const int* src    = (const int*)d_in[9];
  const int* dst    = (const int*)d_in[10];
  const int* psrc   = (const int*)d_in[11];
  const int* pdst   = (const int*)d_in[12];

  char* ws = (char*)d_ws;
  float*    zbuf  = (float*)(ws + 0);              // 50000*64*4 = 12.8 MB
  float*    hbuf  = (float*)(ws + 12800000);       // 12.8 MB
  float*    aggb  = (float*)(ws + 25600000);       // 12.8 MB
  float*    eedge = (float*)(ws + 38400000);       // 3.2 MB
  float*    el    = (float*)(ws + 41600000);       // 200 KB
  float*    er    = (float*)(ws + 41800000);       // 200 KB
  unsigned* emax  = (unsigned*)(ws + 42000000);    // 200 KB
  float*    denom = (float*)(ws + 42200000);       // 200 KB
  unsigned long long* tab = (unsigned long long*)(ws + 42400000); // 16 MB

  // --- build edge-existence hash set (layer-independent) ---
  hipMemsetAsync(tab, 0, (size_t)HASH_CAP * 8, stream);
  hash_insert<<<(NE + 255) / 256, 256, 0, stream>>>(src, dst, tab);

  // --- layer 1: feat (128) -> hbuf (64) ---
  hipMemsetAsync(emax, 0, (size_t)NN * 4, stream);
  hipMemsetAsync(denom, 0, (size_t)NN * 4, stream);
  hipMemsetAsync(aggb, 0, (size_t)NN * 64 * 4, stream);
  gemm_node_wmma<128><<<NN / 16, 128, 0, stream>>>(feat, W1, zbuf);
  elr_kernel<<<(NN * 32) / 256, 256, 0, stream>>>(zbuf, al1, ar1, el, er);
  edge_attn_max<<<(NE + 255) / 256, 256, 0, stream>>>(src, dst, el, er, eedge,
                                                      emax);
  edge_aggregate<<<(NE * 32) / 256, 256, 0, stream>>>(src, dst, eedge, emax,
                                                      zbuf, denom, aggb);
  norm_relu<<<(NN * 64 + 255) / 256, 256, 0, stream>>>(aggb, denom, hbuf);

  // --- layer 2: hbuf (64) -> hbuf (64) ---
  hipMemsetAsync(emax, 0, (size_t)NN * 4, stream);
  hipMemsetAsync(denom, 0, (size_t)NN * 4, stream);
  hipMemsetAsync(aggb, 0, (size_t)NN * 64 * 4, stream);
  gemm_node_wmma<64><<<NN / 16, 128, 0, stream>>>(hbuf, W2, zbuf);
  elr_kernel<<<(NN * 32) / 256, 256, 0, stream>>>(zbuf, al2, ar2, el, er);
  edge_attn_max<<<(NE + 255) / 256, 256, 0, stream>>>(src, dst, el, er, eedge,
                                                      emax);
  edge_aggregate<<<(NE * 32) / 256, 256, 0, stream>>>(src, dst, eedge, emax,
                                                      zbuf, denom, aggb);
  norm_relu<<<(NN * 64 + 255) / 256, 256, 0, stream>>>(aggb, denom, hbuf);

  // --- pair scoring + mask ---
  pair_score<<<(NP * 32) / 256, 256, 0, stream>>>(hbuf, Wc, bc, psrc, pdst,
                                                  tab, (float*)d_out);
}